// GCN_with_feature_64914135712500
// MI455X (gfx1250) — compile-verified
//
#include <hip/hip_runtime.h>

#define D 128
#define LDP 132  // padded LDS row stride (floats); 132*4B = 528B = multiple of 16

typedef __attribute__((ext_vector_type(2))) float v2f;
typedef __attribute__((ext_vector_type(8))) float v8f;

struct __align__(16) F4 { float x, y, z, w; };

__global__ void zero3_kernel(int* a, int* b, int* c, int n) {
  int i = blockIdx.x * blockDim.x + threadIdx.x;
  if (i < n) { a[i] = 0; b[i] = 0; c[i] = 0; }
}

__global__ void degree_kernel(const int* __restrict__ src, const int* __restrict__ dst,
                              int* deg_out, int* deg_in, int e) {
  int i = blockIdx.x * blockDim.x + threadIdx.x;
  if (i < e) {
    atomicAdd(&deg_out[src[i]], 1);
    atomicAdd(&deg_in[dst[i]], 1);
  }
}

__global__ void norm_kernel(const int* __restrict__ deg_out, const int* __restrict__ deg_in,
                            float* __restrict__ dout, float* __restrict__ din, int n) {
  int i = blockIdx.x * blockDim.x + threadIdx.x;
  if (i < n) {
    dout[i] = rsqrtf(fmaxf((float)deg_out[i], 1.0f));
    din[i]  = rsqrtf(fmaxf((float)deg_in[i], 1.0f));
  }
}

// Single-workgroup Hillis-Steele scan over chunks of 1024 with running carry.
__global__ void scan_kernel(const int* __restrict__ deg, int* __restrict__ row_ptr, int n) {
  __shared__ int tmp[1024];
  __shared__ int carry_s;
  int tid = threadIdx.x;
  if (tid == 0) { carry_s = 0; row_ptr[0] = 0; }
  __syncthreads();
  int nchunks = (n + 1023) / 1024;
  for (int c = 0; c < nchunks; ++c) {
    int idx = c * 1024 + tid;
    int v = (idx < n) ? deg[idx] : 0;
    tmp[tid] = v;
    __syncthreads();
    for (int off = 1; off < 1024; off <<= 1) {
      int t = (tid >= off) ? tmp[tid - off] : 0;
      __syncthreads();
      if (tid >= off) tmp[tid] += t;
      __syncthreads();
    }
    int incl = tmp[tid] + carry_s;
    if (idx < n) row_ptr[idx + 1] = incl;
    __syncthreads();
    if (tid == 1023) carry_s = incl;
    __syncthreads();
  }
}

__global__ void fill_kernel(const int* __restrict__ src, const int* __restrict__ dst,
                            const int* __restrict__ row_ptr, int* cursor,
                            int* __restrict__ csr_src, int e) {
  int i = blockIdx.x * blockDim.x + threadIdx.x;
  if (i < e) {
    int d = dst[i];
    int pos = atomicAdd(&cursor[d], 1);
    csr_src[row_ptr[d] + pos] = src[i];
  }
}

// Y[i,:] = dout[i] * (X[i,:] @ W)   via V_WMMA_F32_16X16X4_F32
// 256 threads = 8 waves; wave w owns output column tile [16w, 16w+16).
// Each workgroup handles 8 row-blocks of 16 (128 rows), reusing W staged in LDS.
__global__ void __launch_bounds__(256)
gemm_scale_kernel(const float* __restrict__ X, const float* __restrict__ W,
                  const float* __restrict__ dout, float* __restrict__ Y, int n) {
  __shared__ float Wt[D * LDP];   // transposed: Wt[col][k]
  __shared__ float Xl[16 * LDP];  // Xl[row][k]
  int tid = threadIdx.x;
  for (int idx = tid; idx < D * D; idx += 256) {
    int k = idx >> 7, c = idx & (D - 1);
    Wt[c * LDP + k] = W[idx];     // coalesced global read, strided LDS store
  }
  __syncthreads();
  int wave = tid >> 5, lane = tid & 31;
  int mn   = lane & 15;
  int half = lane >> 4;
  int koff = half << 1;           // lanes 0-15: K 0,1 ; lanes 16-31: K 2,3
  for (int rb = 0; rb < 8; ++rb) {
    int rowbase = (blockIdx.x * 8 + rb) * 16;
    for (int idx = tid; idx < 16 * D; idx += 256) {
      int r = idx >> 7, c = idx & (D - 1);
      int gr = rowbase + r;
      Xl[r * LDP + c] = (gr < n) ? X[(size_t)gr * D + c] : 0.0f;
    }
    __syncthreads();
    v8f acc = {};
#pragma unroll
    for (int ks = 0; ks < 32; ++ks) {
      int k0 = ks * 4 + koff;
      v2f a = *(const v2f*)&Xl[mn * LDP + k0];
      v2f b = *(const v2f*)&Wt[(wave * 16 + mn) * LDP + k0];
      acc = __builtin_amdgcn_wmma_f32_16x16x4_f32(
          /*neg_a=*/false, a, /*neg_b=*/false, b,
          /*c_mod=*/(short)0, acc, /*reuse_a=*/false, /*reuse_b=*/false);
    }
#pragma unroll
    for (int r = 0; r < 8; ++r) {
      int row = rowbase + half * 8 + r;   // C layout: VGPR r -> M=r (lanes<16), M=8+r (lanes>=16)
      if (row < n) {
        Y[(size_t)row * D + wave * 16 + mn] = acc[r] * dout[row];
      }
    }
    __syncthreads();
  }
}

// out[i,:] = din[i] * sum_{j in csr[i]} Y[j,:] + bias
// One wave per node; lane owns a float4 slice (b128 loads, Y is L2-resident).
__global__ void __launch_bounds__(256)
aggregate_kernel(const float* __restrict__ Y, const int* __restrict__ row_ptr,
                 const int* __restrict__ csr_src, const float* __restrict__ din,
                 const float* __restrict__ bias, float* __restrict__ out, int n) {
  int wave = threadIdx.x >> 5;
  int lane = threadIdx.x & 31;
  int node = blockIdx.x * 8 + wave;
  if (node >= n) return;
  int beg = row_ptr[node], end = row_ptr[node + 1];
  const F4* Y4 = (const F4*)Y;
  float sx = 0.f, sy = 0.f, sz = 0.f, sw = 0.f;
  for (int j = beg; j < end; ++j) {
    int s = csr_src[j];
    F4 y = Y4[(size_t)s * 32 + lane];
    sx += y.x; sy += y.y; sz += y.z; sw += y.w;
  }
  float dn = din[node];
  F4 bb = ((const F4*)bias)[lane];
  F4 o;
  o.x = sx * dn + bb.x;
  o.y = sy * dn + bb.y;
  o.z = sz * dn + bb.z;
  o.w = sw * dn + bb.w;
  ((F4*)out)[(size_t)node * 32 + lane] = o;
}

extern "C" void kernel_launch(void* const* d_in, const int* in_sizes, int n_in,
                              void* d_out, int out_size, void* d_ws, size_t ws_size,
                              hipStream_t stream) {
  const float* in_feat = (const float*)d_in[0];
  const int*   edge    = (const int*)d_in[1];
  const float* W1      = (const float*)d_in[2];
  const float* b1      = (const float*)d_in[3];
  const float* W2      = (const float*)d_in[4];
  const float* b2      = (const float*)d_in[5];
  (void)n_in; (void)out_size; (void)ws_size;

  int N = in_sizes[0] / D;
  int E = in_sizes[1] / 2;
  const int* src = edge;
  const int* dst = edge + E;

  char* ws = (char*)d_ws;
  size_t off = 0;
  auto walloc = [&](size_t bytes) {
    char* p = ws + off;
    off += (bytes + 255) & ~(size_t)255;
    return p;
  };
  int*   deg_out = (int*)walloc((size_t)N * 4);
  int*   deg_in  = (int*)walloc((size_t)N * 4);
  int*   cursor  = (int*)walloc((size_t)N * 4);
  int*   row_ptr = (int*)walloc((size_t)(N + 1) * 4);
  float* dout    = (float*)walloc((size_t)N * 4);
  float* din     = (float*)walloc((size_t)N * 4);
  int*   csr     = (int*)walloc((size_t)E * 4);
  float* Y       = (float*)walloc((size_t)N * D * 4);

  const int tb = 256;
  zero3_kernel<<<(N + tb - 1) / tb, tb, 0, stream>>>(deg_out, deg_in, cursor, N);
  degree_kernel<<<(E + tb - 1) / tb, tb, 0, stream>>>(src, dst, deg_out, deg_in, E);
  norm_kernel<<<(N + tb - 1) / tb, tb, 0, stream>>>(deg_out, deg_in, dout, din, N);
  scan_kernel<<<1, 1024, 0, stream>>>(deg_in, row_ptr, N);
  fill_kernel<<<(E + tb - 1) / tb, tb, 0, stream>>>(src, dst, row_ptr, cursor, csr, E);

  float* H = (float*)d_out;
  int gemm_wgs = (N + 127) / 128;
  int agg_wgs  = (N + 7) / 8;

  // Layer 1: H = Adj_norm @ (in_feat @ W1) + b1
  gemm_scale_kernel<<<gemm_wgs, 256, 0, stream>>>(in_feat, W1, dout, Y, N);
  aggregate_kernel<<<agg_wgs, 256, 0, stream>>>(Y, row_ptr, csr, din, b1, H, N);

  // Layer 2: H = Adj_norm @ (H @ W2) + b2
  gemm_scale_kernel<<<gemm_wgs, 256, 0, stream>>>(H, W2, dout, Y, N);
  aggregate_kernel<<<agg_wgs, 256, 0, stream>>>(Y, row_ptr, csr, din, b2, H, N);
}